// MPNNs_60198261620971
// MI455X (gfx1250) — compile-verified
//
#include <hip/hip_runtime.h>
#include <hip/hip_bf16.h>
#include <math.h>

// ---------------------------------------------------------------------------
// GCN forward on MI455X (gfx1250, wave32).
// - GEMMs: V_WMMA_F32_16X16X4_F32 (fp32-exact; memory-bound problem, so no
//   reward for reduced precision).
// - Weights pre-transposed once to workspace, then staged per-block into LDS
//   by the Tensor Data Mover (tensor_load_to_lds) with pad_enable producing a
//   padded [n][k] layout: every WMMA B-fragment is a single aligned
//   ds_load_b64 (no shuffle moves, conflict-free banks).
// - Edge aggregation: coalesced L2 fp32 atomics.
// ---------------------------------------------------------------------------

#ifndef USE_TDM
#define USE_TDM 1
#endif

typedef float v2f __attribute__((ext_vector_type(2)));
typedef float v8f __attribute__((ext_vector_type(8)));
typedef unsigned v4u __attribute__((ext_vector_type(4)));
typedef int v4i __attribute__((ext_vector_type(4)));
typedef int v8i __attribute__((ext_vector_type(8)));

#define NN_IN  128
#define NN_HID 64
#define NN_OUT 40
#define KP_IN  (NN_IN + 4)    // 132: padded LDS row stride (dwords) for K=128
#define KP_HID (NN_HID + 4)   // 68:  padded LDS row stride (dwords) for K=64

__device__ __forceinline__ v8f wmma_f32_k4(v2f a, v2f b, v8f c) {
    // D = A(16x4,f32) * B(4x16,f32) + C(16x16,f32)
    return __builtin_amdgcn_wmma_f32_16x16x4_f32(
        /*neg_a=*/false, a, /*neg_b=*/false, b,
        /*c_mod=*/(short)0, c, /*reuse_a=*/false, /*reuse_b=*/false);
}

__device__ __forceinline__ void atomic_add_f32(float* p, float v) {
    __hip_atomic_fetch_add(p, v, __ATOMIC_RELAXED, __HIP_MEMORY_SCOPE_AGENT);
}

// ---------------------------------------------------------------------------
// Tensor Data Mover: 2D tile (row-major, 4-byte elements) global -> LDS with
// optional LDS padding (pad_amount dwords every 2^pad_interval_log2 dwords).
// Descriptor layout per CDNA5 ISA.
// This toolchain uses the 6-arg builtin:
//   (uint32x4 g0, int32x8 g1, int32x4 g2, int32x4 g3, int32x8 g4, i32 cpol)
// ---------------------------------------------------------------------------
__device__ __forceinline__ void tdm_load_2d(unsigned lds_byte_off, const void* gptr,
                                            unsigned dim0, unsigned dim1,
                                            unsigned tile0, unsigned tile1,
                                            unsigned stride0,
                                            unsigned pad_interval_log2,
                                            unsigned pad_amount_dwords) {
#if USE_TDM
    unsigned long long ga = (unsigned long long)(uintptr_t)gptr;
    v4u g0;
    g0[0] = 1u;                                   // count=1, user descriptor
    g0[1] = lds_byte_off;                         // lds_addr
    g0[2] = (unsigned)(ga & 0xFFFFFFFFu);         // global_addr[31:0]
    g0[3] = (unsigned)((ga >> 32) & 0x1FFFFFFu)   // global_addr[56:32]
          | (2u << 30);                           // type = 2 ("image")
    unsigned pe = pad_amount_dwords ? 1u : 0u;
    unsigned pi = pe ? (pad_interval_log2 - 1u) : 0u;  // interval = 2<<pi dwords
    unsigned pa = pe ? (pad_amount_dwords - 1u) : 0u;  // amount = pa+1 dwords
    v8i g1;
    g1[0] = (int)((2u << 16) | (pe << 20) | (pi << 22) | (pa << 25)); // data_size=4B
    g1[1] = (int)((dim0 & 0xFFFFu) << 16);                // tensor_dim0[15:0]
    g1[2] = (int)((dim0 >> 16) | ((dim1 & 0xFFFFu) << 16));
    g1[3] = (int)((dim1 >> 16) | (tile0 << 16));          // tile_dim0
    g1[4] = (int)(tile1 & 0xFFFFu);                       // tile_dim1 (tile_dim2=0)
    g1[5] = (int)stride0;                                 // tensor_dim0_stride[31:0]
    g1[6] = 0;
    g1[7] = 0;
    v4i z4 = {0, 0, 0, 0};
    v8i z8 = {0, 0, 0, 0, 0, 0, 0, 0};
    __builtin_amdgcn_tensor_load_to_lds(g0, g1, z4, z4, z8, 0);
#endif
}

__device__ __forceinline__ void tdm_wait0() {
#if USE_TDM
    __builtin_amdgcn_s_wait_tensorcnt(0);
#endif
}

// Fallback cooperative stage of transposed weight WT[Nn][K] into padded LDS.
__device__ __forceinline__ void coop_stage(float* sW, const float* WT,
                                           int K, int KPad, int Nn, int tid) {
#if !USE_TDM
    for (int i = tid; i < Nn * K; i += 256) {
        int n = i / K, k = i % K;
        sW[n * KPad + k] = WT[i];
    }
#endif
}

// ---------------------------------------------------------------------------
// small elementwise / degree / transpose kernels
// ---------------------------------------------------------------------------

__global__ void k_fill1(float* deg, int n) {
    int i = blockIdx.x * blockDim.x + threadIdx.x;
    if (i < n) deg[i] = 1.0f;  // self loop
}

__global__ void k_deg(const int* __restrict__ dst, float* deg, int E) {
    int e = blockIdx.x * blockDim.x + threadIdx.x;
    if (e < E) atomic_add_f32(&deg[dst[e]], 1.0f);
}

__global__ void k_rsqrt_inplace(float* d, int n) {
    int i = blockIdx.x * blockDim.x + threadIdx.x;
    if (i < n) d[i] = rsqrtf(d[i]);
}

__global__ void k_coef(const int* __restrict__ src, const int* __restrict__ dst,
                       const float* __restrict__ dinv, float* __restrict__ coef, int E) {
    int e = blockIdx.x * blockDim.x + threadIdx.x;
    if (e < E) coef[e] = dinv[src[e]] * dinv[dst[e]];
}

// WT[n*K + k] = W[k*Nn + n]
__global__ void k_transpose(const float* __restrict__ W, float* __restrict__ WT,
                            int K, int Nn) {
    int i = blockIdx.x * blockDim.x + threadIdx.x;
    if (i < K * Nn) {
        int k = i / Nn, n = i % Nn;
        WT[n * K + k] = W[i];
    }
}

// WT[n*K + k] = (n < Nn) ? W[k*Nn + n] : 0   for n in [0, Np)
__global__ void k_transpose_pad(const float* __restrict__ W, float* __restrict__ WT,
                                int K, int Nn, int Np) {
    int i = blockIdx.x * blockDim.x + threadIdx.x;
    if (i < K * Np) {
        int k = i / Np, n = i % Np;
        WT[n * K + k] = (n < Nn) ? W[k * Nn + n] : 0.0f;
    }
}

// ---------------------------------------------------------------------------
// GEMM 1: H[N,64] = X[N,128] @ Wf[128,64] + bf   (WfT pre-transposed [64][128])
// block = 256 threads = 8 waves; each wave -> 16 rows x 64 cols.
// ---------------------------------------------------------------------------

__global__ __launch_bounds__(256) void k_gemm_in(const float* __restrict__ X,
                                                 const float* __restrict__ WfT,
                                                 const float* __restrict__ bf,
                                                 float* __restrict__ H, int N) {
    __shared__ float sW[NN_HID * KP_IN];  // [n][k], padded stride 132
    int tid = threadIdx.x;
    if (tid < 32) {
        tdm_load_2d((unsigned)(uintptr_t)sW, WfT,
                    /*dim0=*/NN_IN, /*dim1=*/NN_HID,
                    /*tile0=*/NN_IN, /*tile1=*/NN_HID,
                    /*stride0=*/NN_IN, /*pad 4 dw every 128 dw*/ 7, 4);
        tdm_wait0();
    }
    coop_stage(sW, WfT, NN_IN, KP_IN, NN_HID, tid);
    __syncthreads();

    int lane = tid & 31, wave = tid >> 5;
    int row0 = (blockIdx.x * 8 + wave) * 16;
    if (row0 >= N) return;

    int mload = row0 + (lane & 15); if (mload >= N) mload = N - 1;
    int koff  = (lane >> 4) * 2;
    int ncol  = lane & 15;
    const float* arow = X + (size_t)mload * NN_IN;

    v8f acc[4] = {};
    for (int kb = 0; kb < NN_IN; kb += 4) {
        v2f a = *(const v2f*)(arow + kb + koff);
#pragma unroll
        for (int t = 0; t < 4; ++t) {
            v2f b = *(const v2f*)&sW[(t * 16 + ncol) * KP_IN + kb + koff];
            acc[t] = wmma_f32_k4(a, b, acc[t]);
        }
    }

    int halfm = (lane >> 4) * 8;
#pragma unroll
    for (int t = 0; t < 4; ++t) {
        int n = t * 16 + ncol;
        float bias = bf[n];
#pragma unroll
        for (int v = 0; v < 8; ++v) {
            int m = row0 + halfm + v;
            if (m < N) H[(size_t)m * NN_HID + n] = acc[t][v] + bias;
        }
    }
}

// ---------------------------------------------------------------------------
// Fused dual GEMM per layer (WcT / WlT pre-transposed [64][64]):
//   HW   = H @ Wc
//   BASE = HW * dinv^2 + bc + H @ Wl + bl     (self-loop term + residual lin)
// ---------------------------------------------------------------------------

__global__ __launch_bounds__(256) void k_gemm_dual(const float* __restrict__ H,
                                                   const float* __restrict__ WcT,
                                                   const float* __restrict__ WlT,
                                                   const float* __restrict__ bc,
                                                   const float* __restrict__ bl,
                                                   const float* __restrict__ dinv,
                                                   float* __restrict__ HW,
                                                   float* __restrict__ BASE, int N) {
    __shared__ float sC[NN_HID * KP_HID];  // [n][k], padded stride 68
    __shared__ float sL[NN_HID * KP_HID];
    int tid = threadIdx.x;
    if (tid < 32) {
        tdm_load_2d((unsigned)(uintptr_t)sC, WcT, NN_HID, NN_HID, NN_HID, NN_HID,
                    NN_HID, /*pad 4 dw every 64 dw*/ 6, 4);
        tdm_load_2d((unsigned)(uintptr_t)sL, WlT, NN_HID, NN_HID, NN_HID, NN_HID,
                    NN_HID, 6, 4);
        tdm_wait0();
    }
    coop_stage(sC, WcT, NN_HID, KP_HID, NN_HID, tid);
    coop_stage(sL, WlT, NN_HID, KP_HID, NN_HID, tid);
    __syncthreads();

    int lane = tid & 31, wave = tid >> 5;
    int row0 = (blockIdx.x * 8 + wave) * 16;
    if (row0 >= N) return;

    int mload = row0 + (lane & 15); if (mload >= N) mload = N - 1;
    int koff  = (lane >> 4) * 2;
    int ncol  = lane & 15;
    const float* arow = H + (size_t)mload * NN_HID;

    v8f accC[4] = {};
    v8f accL[4] = {};
    for (int kb = 0; kb < NN_HID; kb += 4) {
        v2f a = *(const v2f*)(arow + kb + koff);
#pragma unroll
        for (int t = 0; t < 4; ++t) {
            int i0 = (t * 16 + ncol) * KP_HID + kb + koff;
            v2f b0 = *(const v2f*)&sC[i0];
            accC[t] = wmma_f32_k4(a, b0, accC[t]);
            v2f b1 = *(const v2f*)&sL[i0];
            accL[t] = wmma_f32_k4(a, b1, accL[t]);
        }
    }

    int halfm = (lane >> 4) * 8;
#pragma unroll
    for (int t = 0; t < 4; ++t) {
        int n = t * 16 + ncol;
        float bcn = bc[n], bln = bl[n];
#pragma unroll
        for (int v = 0; v < 8; ++v) {
            int m = row0 + halfm + v;
            if (m < N) {
                float hwv = accC[t][v];
                float di  = dinv[m];
                size_t off = (size_t)m * NN_HID + n;
                HW[off]   = hwv;
                BASE[off] = hwv * di * di + bcn + accL[t][v] + bln;
            }
        }
    }
}

// ---------------------------------------------------------------------------
// Edge scatter: BASE[dst, c] += coef[e] * HW[src, c]
// 64 lanes per edge -> contiguous 256B gather + contiguous 256B L2 atomics.
// ---------------------------------------------------------------------------

__global__ __launch_bounds__(256) void k_scatter(const int* __restrict__ src,
                                                 const int* __restrict__ dst,
                                                 const float* __restrict__ coef,
                                                 const float* __restrict__ HW,
                                                 float* __restrict__ BASE, int E) {
    unsigned idx = blockIdx.x * 256u + threadIdx.x;
    unsigned e = idx >> 6;
    unsigned c = idx & 63u;
    if (e < (unsigned)E) {
        float v = coef[e] * HW[(size_t)src[e] * NN_HID + c];
        atomic_add_f32(&BASE[(size_t)dst[e] * NN_HID + c], v);
    }
}

// ---------------------------------------------------------------------------
// BatchNorm (eval) + ReLU:  H = relu((BASE - mean) * gamma*rsqrt(var+eps) + beta)
// ---------------------------------------------------------------------------

__global__ __launch_bounds__(256) void k_bnrelu(const float* __restrict__ BASE,
                                                const float* __restrict__ mean,
                                                const float* __restrict__ var,
                                                const float* __restrict__ gamma,
                                                const float* __restrict__ beta,
                                                float* __restrict__ H, unsigned n64) {
    unsigned i = blockIdx.x * 256u + threadIdx.x;
    if (i < n64) {
        unsigned c = i & 63u;
        float s = gamma[c] * rsqrtf(var[c] + 1e-5f);
        float h = (BASE[i] - mean[c]) * s + beta[c];
        H[i] = h > 0.0f ? h : 0.0f;
    }
}

// ---------------------------------------------------------------------------
// Output head: OUT[N,40] = H[N,64] @ Wp[64,40] + bp
// WpT pre-transposed+padded [48][64]; 3 n-tiles, tail tile stores 8 cols.
// ---------------------------------------------------------------------------

__global__ __launch_bounds__(256) void k_gemm_out(const float* __restrict__ H,
                                                  const float* __restrict__ WpT,
                                                  const float* __restrict__ bp,
                                                  float* __restrict__ OUT, int N) {
    __shared__ float sW[48 * KP_HID];  // [n][k], padded stride 68
    int tid = threadIdx.x;
    if (tid < 32) {
        tdm_load_2d((unsigned)(uintptr_t)sW, WpT, NN_HID, 48, NN_HID, 48,
                    NN_HID, 6, 4);
        tdm_wait0();
    }
    coop_stage(sW, WpT, NN_HID, KP_HID, 48, tid);
    __syncthreads();

    int lane = tid & 31, wave = tid >> 5;
    int row0 = (blockIdx.x * 8 + wave) * 16;
    if (row0 >= N) return;

    int mload = row0 + (lane & 15); if (mload >= N) mload = N - 1;
    int koff  = (lane >> 4) * 2;
    int ncol  = lane & 15;
    const float* arow = H + (size_t)mload * NN_HID;

    v8f acc[3] = {};
    for (int kb = 0; kb < NN_HID; kb += 4) {
        v2f a = *(const v2f*)(arow + kb + koff);
#pragma unroll
        for (int t = 0; t < 3; ++t) {
            v2f b = *(const v2f*)&sW[(t * 16 + ncol) * KP_HID + kb + koff];
            acc[t] = wmma_f32_k4(a, b, acc[t]);
        }
    }

    int halfm = (lane >> 4) * 8;
#pragma unroll
    for (int t = 0; t < 3; ++t) {
        int n = t * 16 + ncol;
        if (n >= NN_OUT) continue;
        float bias = bp[n];
#pragma unroll
        for (int v = 0; v < 8; ++v) {
            int m = row0 + halfm + v;
            if (m < N) OUT[(size_t)m * NN_OUT + n] = acc[t][v] + bias;
        }
    }
}

// ---------------------------------------------------------------------------
// launcher
// ---------------------------------------------------------------------------

extern "C" void kernel_launch(void* const* d_in, const int* in_sizes, int n_in,
                              void* d_out, int out_size, void* d_ws, size_t ws_size,
                              hipStream_t stream) {
    const float* x     = (const float*)d_in[0];
    const int*   ei    = (const int*)  d_in[1];
    const float* Wf    = (const float*)d_in[2];
    const float* bf    = (const float*)d_in[3];
    const float* convW = (const float*)d_in[4];
    const float* convB = (const float*)d_in[5];
    const float* linW  = (const float*)d_in[6];
    const float* linB  = (const float*)d_in[7];
    const float* gamma = (const float*)d_in[8];
    const float* beta  = (const float*)d_in[9];
    const float* rmean = (const float*)d_in[10];
    const float* rvar  = (const float*)d_in[11];
    const float* predW = (const float*)d_in[12];
    const float* predB = (const float*)d_in[13];
    float* out = (float*)d_out;

    const int N = in_sizes[0] / NN_IN;
    const int E = in_sizes[1] / 2;
    const int* src = ei;
    const int* dst = ei + E;

    float* ws    = (float*)d_ws;
    float* dinv  = ws;                          // N
    float* coef  = dinv + N;                    // E
    float* H     = coef + E;                    // N*64
    float* HW    = H     + (size_t)N * NN_HID;  // N*64
    float* BASE  = HW    + (size_t)N * NN_HID;  // N*64
    float* WfT   = BASE  + (size_t)N * NN_HID;  // 64*128
    float* convT = WfT   + NN_HID * NN_IN;      // 3 * 64*64
    float* linT  = convT + 3 * NN_HID * NN_HID; // 3 * 64*64
    float* predT = linT  + 3 * NN_HID * NN_HID; // 48*64

    const int t256n = (N + 255) / 256;
    const int t256e = (E + 255) / 256;
    const int gblk  = (N + 127) / 128;  // 8 waves x 16 rows per block

    // one-off weight transposes (feed the TDM-staged padded LDS layout)
    k_transpose<<<(NN_IN * NN_HID + 255) / 256, 256, 0, stream>>>(Wf, WfT, NN_IN, NN_HID);
    for (int l = 0; l < 3; ++l) {
        k_transpose<<<(NN_HID * NN_HID + 255) / 256, 256, 0, stream>>>(
            convW + l * NN_HID * NN_HID, convT + l * NN_HID * NN_HID, NN_HID, NN_HID);
        k_transpose<<<(NN_HID * NN_HID + 255) / 256, 256, 0, stream>>>(
            linW + l * NN_HID * NN_HID, linT + l * NN_HID * NN_HID, NN_HID, NN_HID);
    }
    k_transpose_pad<<<(NN_HID * 48 + 255) / 256, 256, 0, stream>>>(predW, predT,
                                                                   NN_HID, NN_OUT, 48);

    // degree -> dinv -> edge coefficients (computed once; reused per layer)
    k_fill1        <<<t256n, 256, 0, stream>>>(dinv, N);
    k_deg          <<<t256e, 256, 0, stream>>>(dst, dinv, E);
    k_rsqrt_inplace<<<t256n, 256, 0, stream>>>(dinv, N);
    k_coef         <<<t256e, 256, 0, stream>>>(src, dst, dinv, coef, E);

    // input projection
    k_gemm_in<<<gblk, 256, 0, stream>>>(x, WfT, bf, H, N);

    const unsigned n64 = (unsigned)N * NN_HID;
    for (int l = 0; l < 3; ++l) {
        k_gemm_dual<<<gblk, 256, 0, stream>>>(H,
                                              convT + l * NN_HID * NN_HID,
                                              linT  + l * NN_HID * NN_HID,
                                              convB + l * NN_HID,
                                              linB  + l * NN_HID,
                                              dinv, HW, BASE, N);
        unsigned total = (unsigned)E * 64u;
        k_scatter<<<(total + 255) / 256, 256, 0, stream>>>(src, dst, coef, HW, BASE, E);
        k_bnrelu<<<(n64 + 255) / 256, 256, 0, stream>>>(BASE,
                                                        rmean + l * NN_HID,
                                                        rvar  + l * NN_HID,
                                                        gamma + l * NN_HID,
                                                        beta  + l * NN_HID,
                                                        H, n64);
    }

    // output head
    k_gemm_out<<<gblk, 256, 0, stream>>>(H, predT, predB, out, N);
}